// BaseSampler_58025008169306
// MI455X (gfx1250) — compile-verified
//
#include <hip/hip_runtime.h>
#include <hip/hip_bf16.h>
#include <stdint.h>

// Problem constants (match the reference setup; VOCAB is compile-time for LDS sizing)
#define VOCAB      32000
#define VCHUNKS    (VOCAB / 4)      // 8000 x b128 chunks per row
#define NBIN       4096             // radix-select histogram bins (top 12 key bits)
#define BLK        256
#define NWAVE      (BLK / 32)
#define BINS_PER_T (NBIN / BLK)     // 16
#define CBUF       1024             // tail-collection buffer (top-k tail is ~50-60 elems)

// ---------------- CDNA5 async / tensor global -> LDS paths ------------------
typedef int v4i __attribute__((ext_vector_type(4)));
typedef __attribute__((address_space(1))) v4i* global_v4i_ptr;
typedef __attribute__((address_space(3))) v4i* lds_v4i_ptr;

typedef unsigned int u32x4 __attribute__((ext_vector_type(4)));
typedef int          i32x4 __attribute__((ext_vector_type(4)));
typedef int          i32x8 __attribute__((ext_vector_type(8)));

#if defined(__gfx1250__) && __has_builtin(__builtin_amdgcn_tensor_load_to_lds)
#define HAVE_TDM 1
#else
#define HAVE_TDM 0
#endif

__device__ __forceinline__ void async_copy_f4(const float* gsrc, float* lds_dst) {
#if defined(__gfx1250__) && __has_builtin(__builtin_amdgcn_global_load_async_to_lds_b128)
    __builtin_amdgcn_global_load_async_to_lds_b128(
        (global_v4i_ptr)(gsrc),
        (lds_v4i_ptr)(lds_dst),
        0, 0);
#else
    *(float4*)lds_dst = *(const float4*)gsrc;   // sync fallback (still correct)
#endif
}

__device__ __forceinline__ void wait_async_zero() {
#if defined(__gfx1250__) && __has_builtin(__builtin_amdgcn_s_wait_asynccnt)
    __builtin_amdgcn_s_wait_asynccnt(0);
#elif defined(__gfx1250__)
    asm volatile("s_wait_asynccnt 0" ::: "memory");
#endif
}

#if HAVE_TDM
// One TDM descriptor DMAs the whole 32000-float row (2D tile: 32000 x 1, 4B
// elements) into LDS. D# layout per CDNA5 ISA ch.8 (async_tensor doc).
__device__ __forceinline__ void tdm_load_row(const float* gsrc, unsigned lds_byte_addr) {
    const unsigned long long ga = (unsigned long long)(uintptr_t)gsrc;
    u32x4 g0;
    g0.x = 1u;                                            // count=1, user mode
    g0.y = lds_byte_addr;                                 // lds_addr[63:32]
    g0.z = (unsigned)(ga & 0xFFFFFFFFull);                // global_addr[31:0]
    g0.w = (unsigned)((ga >> 32) & 0x01FFFFFFull)         // global_addr[56:32]
           | 0x80000000u;                                 // type=2 ("image")
    i32x8 g1;
    g1[0] = (int)(2u << 16);                              // data_size=4B; mask/flags=0
    g1[1] = (int)(((unsigned)VOCAB & 0xFFFFu) << 16);     // tensor_dim0[15:0] @bits[63:48]
    g1[2] = (int)((((unsigned)VOCAB >> 16) & 0xFFFFu)     // tensor_dim0[31:16]
                  | (1u << 16));                          // tensor_dim1[15:0] = 1
    g1[3] = (int)(((unsigned)VOCAB) << 16);               // tensor_dim1[31:16]=0 | tile_dim0
    g1[4] = 1;                                            // tile_dim1=1, tile_dim2=0
    g1[5] = VOCAB;                                        // tensor_dim0_stride[31:0]
    g1[6] = 0;                                            // stride0[47:32] | stride1[15:0]
    g1[7] = 0;                                            // stride1[47:16]
    const i32x4 gz = {0, 0, 0, 0};
#if __has_include(<hip/amd_detail/amd_gfx1250_TDM.h>)
    const i32x8 gz8 = {0, 0, 0, 0, 0, 0, 0, 0};
    __builtin_amdgcn_tensor_load_to_lds(g0, g1, gz, gz, gz8, 0);   // clang-23 form
#else
    __builtin_amdgcn_tensor_load_to_lds(g0, g1, gz, gz, 0);        // ROCm 7.2 form
#endif
}
#endif

// order-preserving float -> uint key (monotone: larger float => larger key)
__device__ __forceinline__ unsigned sortkey(float x) {
    unsigned u = __float_as_uint(x);
    return u ^ (((unsigned)((int)u >> 31)) | 0x80000000u);
}

// ============================================================================
// Kernel A: one workgroup per candidate token. TDM-streams the 125KB logits
// row into LDS, then computes max/argmax, rank-50 threshold via histogram
// radix-select (DS atomics), collects + sorts the tail, and emits
// top_prob = 1/Z' and the greedy token for this candidate.
// ============================================================================
__global__ void __launch_bounds__(BLK)
sampler_row_kernel(const float* __restrict__ logits,
                   const int*   __restrict__ relative_idx,
                   const int*   __restrict__ cu_filtered,
                   const int*   __restrict__ cu_seqlens_q,
                   const float* __restrict__ temperatures,
                   const float* __restrict__ top_p,
                   const int*   __restrict__ top_k,
                   float*       __restrict__ ws_probs,
                   int*         __restrict__ ws_tokens,
                   int bsz)
{
    extern __shared__ char smem_raw[];
    float*    row  = (float*)smem_raw;                                  // 128000 B
    unsigned* hist = (unsigned*)(smem_raw + (size_t)VOCAB * 4);         //  16384 B
    float*    cbuf = (float*)(smem_raw + (size_t)VOCAB * 4 + NBIN * 4); //   4096 B

    __shared__ float    s_wv[NWAVE];
    __shared__ int      s_wi[NWAVE];
    __shared__ unsigned s_chunk[BLK];
    __shared__ float    s_maxv;
    __shared__ int      s_maxi;
    __shared__ unsigned s_thrfloor;
    __shared__ unsigned s_cnt;

    const int t   = blockIdx.x;
    const int tid = threadIdx.x;

    // group id via binary search in cu_filtered (uniform across block -> SALU)
    int lo = 0, hi = bsz;
    while (lo + 1 < hi) {
        int mid = (lo + hi) >> 1;
        if (cu_filtered[mid] <= t) lo = mid; else hi = mid;
    }
    const int g    = lo;
    const int srow = cu_seqlens_q[g] + relative_idx[t];
    const float invT = 1.0f / temperatures[t];
    const float* gsrc = logits + (size_t)srow * VOCAB;

    // -------- stage 1: DMA the full row into LDS ----------------------------
#if HAVE_TDM
    if (tid < 32) {  // wave 0 issues the single tensor DMA descriptor
        const unsigned lds_row =
            (unsigned)(uintptr_t)(__attribute__((address_space(3))) char*)smem_raw;
        tdm_load_row(gsrc, lds_row);
    }
    // zero the histogram while the TDM transfer is in flight
    for (int i = tid; i < NBIN; i += BLK) hist[i] = 0u;
    if (tid < 32) __builtin_amdgcn_s_wait_tensorcnt(0);
#else
    for (int i = tid; i < VCHUNKS; i += BLK)
        async_copy_f4(gsrc + 4 * i, row + 4 * i);
    for (int i = tid; i < NBIN; i += BLK) hist[i] = 0u;
    wait_async_zero();
#endif
    __syncthreads();

    // -------- stage 2: max + argmax (first-index tiebreak) ------------------
    float m = -__builtin_inff(); int mi = 0;
    for (int v = tid; v < VOCAB; v += BLK) {
        float x = row[v];
        if (x > m) { m = x; mi = v; }
    }
    for (int off = 16; off > 0; off >>= 1) {
        float om = __shfl_down(m, off);
        int   oi = __shfl_down(mi, off);
        if (om > m || (om == m && oi < mi)) { m = om; mi = oi; }
    }
    if ((tid & 31) == 0) { s_wv[tid >> 5] = m; s_wi[tid >> 5] = mi; }
    __syncthreads();
    if (tid == 0) {
        m = s_wv[0]; mi = s_wi[0];
        for (int w = 1; w < NWAVE; ++w)
            if (s_wv[w] > m || (s_wv[w] == m && s_wi[w] < mi)) { m = s_wv[w]; mi = s_wi[w]; }
        s_maxv = m; s_maxi = mi;
    }
    __syncthreads();

    // -------- stage 3: radix-select rank-k floor via 4096-bin histogram -----
    for (int v = tid; v < VOCAB; v += BLK)
        atomicAdd(&hist[sortkey(row[v]) >> 20], 1u);
    __syncthreads();
    unsigned csum = 0;
    for (int i = 0; i < BINS_PER_T; ++i) csum += hist[tid * BINS_PER_T + i];
    s_chunk[tid] = csum;
    __syncthreads();
    if (tid == 0) {
        int kk = top_k[0]; if (kk > VOCAB) kk = VOCAB; if (kk < 1) kk = 1;
        unsigned acc = 0; int c = BLK - 1;
        for (; c > 0; --c) { if (acc + s_chunk[c] >= (unsigned)kk) break; acc += s_chunk[c]; }
        int b = c * BINS_PER_T + BINS_PER_T - 1;
        for (; b > c * BINS_PER_T; --b) { if (acc + hist[b] >= (unsigned)kk) break; acc += hist[b]; }
        s_thrfloor = ((unsigned)b) << 20;
        s_cnt = 0u;
    }
    __syncthreads();

    // -------- stage 4: collect tail values >= bin floor ---------------------
    const unsigned thrf = s_thrfloor;
    for (int v = tid; v < VOCAB; v += BLK) {
        float x = row[v];
        if (sortkey(x) >= thrf) {
            unsigned slot = atomicAdd(&s_cnt, 1u);
            if (slot < CBUF) cbuf[slot] = x;
        }
    }
    __syncthreads();
    const int M = (s_cnt < (unsigned)CBUF) ? (int)s_cnt : CBUF;
    for (int i = tid; i < CBUF; i += BLK)
        if (i >= M) cbuf[i] = -__builtin_inff();
    __syncthreads();

    // -------- stage 5: bitonic sort the buffer, descending ------------------
    for (int k = 2; k <= CBUF; k <<= 1) {
        for (int j = k >> 1; j > 0; j >>= 1) {
            for (int i = tid; i < CBUF; i += BLK) {
                int ixj = i ^ j;
                if (ixj > i) {
                    float a = cbuf[i], b2 = cbuf[ixj];
                    bool descBlk = ((i & k) == 0);
                    bool doswap  = descBlk ? (a < b2) : (a > b2);
                    if (doswap) { cbuf[i] = b2; cbuf[ixj] = a; }
                }
            }
            __syncthreads();
        }
    }

    // -------- stage 6: serial top-k / top-p math on <= ~64 elements ---------
    if (tid == 0) {
        float prob = 1.0f;
        if (M > 0) {
            int kk = top_k[0]; if (kk > VOCAB) kk = VOCAB; if (kk < 1) kk = 1;
            int J = (kk < M) ? kk : M;
            float thr = cbuf[J - 1];
            while (J < M && cbuf[J] >= thr) ++J;   // top-k keeps >= thr (ties)
            const float mx = s_maxv;
            const float p  = top_p[0];
            float Z50 = 0.0f;
            for (int j = 0; j < J; ++j) Z50 += __expf((cbuf[j] - mx) * invT);
            float cum = 0.0f, Zp = 0.0f;
            for (int j = 0; j < J; ++j) {
                float ej = __expf((cbuf[j] - mx) * invT);
                // keep element j iff (normalized) cumsum of previous <= top_p
                if (j == 0 || cum <= p * Z50) Zp += ej;
                cum += ej;
            }
            prob = 1.0f / Zp;                     // max prob = exp(0)/Z'
        }
        ws_probs[t]  = prob;
        ws_tokens[t] = s_maxi;                    // argmax survives both masks
    }
}

// ============================================================================
// Kernel B: one wave32 per batch. Register bitonic sort of the 32 candidates
// by score (descending), then filter by num_transfer / threshold.
// ============================================================================
__global__ void __launch_bounds__(32)
finalize_kernel(const int*   __restrict__ relative_idx,
                const int*   __restrict__ batch_offsets,
                const int*   __restrict__ cu_filtered,
                const int*   __restrict__ num_transfer,
                const float* __restrict__ thresholds,
                const float* __restrict__ ws_probs,
                const int*   __restrict__ ws_tokens,
                int*         __restrict__ out,        // [pos | tokens | counts]
                const int*   __restrict__ block_size_p,
                int bsz)
{
    const int b  = blockIdx.x;
    const int l  = threadIdx.x;
    const int bs = block_size_p[0];
    const int start = cu_filtered[b];
    const int cnt   = cu_filtered[b + 1] - start;

    const bool valid = (l < bs) && (l < cnt);
    float score = valid ? ws_probs[start + l] : -__builtin_inff();
    int   tok   = valid ? ws_tokens[start + l] : 0;
    int   pos   = valid ? (relative_idx[start + l] + batch_offsets[b]) : 0;

    // wave32 bitonic sort, descending by score, carrying (tok, pos)
    for (int k = 2; k <= 32; k <<= 1) {
        for (int j = k >> 1; j > 0; j >>= 1) {
            float os = __shfl_xor(score, j);
            int   ot = __shfl_xor(tok, j);
            int   op = __shfl_xor(pos, j);
            bool lower   = ((l & j) == 0);
            bool descBlk = ((l & k) == 0);
            bool takeMax = (lower == descBlk);
            bool pick    = takeMax ? (os > score) : (os < score);
            if (pick) { score = os; tok = ot; pos = op; }
        }
    }

    int kv = num_transfer[b]; if (kv < 0) kv = 0;
    const float th = thresholds[b];
    const bool keep = (l < kv) && (score >= th) && (score > -__builtin_inff());
    const unsigned long long ball = __ballot(keep ? 1 : 0);

    int* out_pos = out;
    int* out_tok = out + (size_t)bsz * bs;
    int* out_cnt = out + 2 * (size_t)bsz * bs;
    if (l < bs) {
        out_pos[b * bs + l] = keep ? pos : 0;
        out_tok[b * bs + l] = keep ? tok : -1;
    }
    if (l == 0) out_cnt[b] = __popcll(ball);
}

// ============================================================================
extern "C" void kernel_launch(void* const* d_in, const int* in_sizes, int n_in,
                              void* d_out, int out_size, void* d_ws, size_t ws_size,
                              hipStream_t stream) {
    (void)n_in; (void)out_size; (void)ws_size;

    const float* logits        = (const float*)d_in[0];
    const int*   relative_idx  = (const int*)  d_in[1];
    const int*   batch_offsets = (const int*)  d_in[2];
    const int*   cu_filtered   = (const int*)  d_in[3];
    const int*   cu_seqlens_q  = (const int*)  d_in[4];
    const float* temperatures  = (const float*)d_in[5];
    const int*   num_transfer  = (const int*)  d_in[6];
    const float* thresholds    = (const float*)d_in[7];
    const float* top_p         = (const float*)d_in[8];
    const int*   top_k         = (const int*)  d_in[9];
    const int*   block_size    = (const int*)  d_in[10];

    const int total = in_sizes[1];       // candidate tokens
    const int bsz   = in_sizes[3] - 1;   // batches

    float* ws_probs  = (float*)d_ws;
    int*   ws_tokens = (int*)((char*)d_ws + sizeof(float) * (size_t)total);

    const size_t smem = (size_t)VOCAB * 4 + (size_t)NBIN * 4 + (size_t)CBUF * 4;

    sampler_row_kernel<<<total, BLK, smem, stream>>>(
        logits, relative_idx, cu_filtered, cu_seqlens_q, temperatures,
        top_p, top_k, ws_probs, ws_tokens, bsz);

    finalize_kernel<<<bsz, 32, 0, stream>>>(
        relative_idx, batch_offsets, cu_filtered, num_transfer, thresholds,
        ws_probs, ws_tokens, (int*)d_out, block_size, bsz);
}